// ImprovedMolGraphTransformer_16544214024630
// MI455X (gfx1250) — compile-verified
//
#include <hip/hip_runtime.h>
#include <math.h>

#define HID 128
#define HEADS 4
#define DH 32
#define NGRAPH 512

typedef __attribute__((ext_vector_type(2))) float v2f;
typedef __attribute__((ext_vector_type(8))) float v8f;

__device__ __forceinline__ float waveSum(float v) {
#pragma unroll
  for (int off = 16; off > 0; off >>= 1) v += __shfl_xor(v, off, 32);
  return v;
}

__device__ __forceinline__ void atomicMaxF(float* addr, float val) {
  int* ia = reinterpret_cast<int*>(addr);
  int cur = __float_as_int(*addr);
  while (val > __int_as_float(cur)) {
    int old = atomicCAS(ia, cur, __float_as_int(val));
    if (old == cur) break;
    cur = old;
  }
}

// ---------------- fill ----------------
__global__ void fill_kernel(float* __restrict__ p, float v, long long n) {
  long long i = (long long)blockIdx.x * blockDim.x + threadIdx.x;
  long long stride = (long long)gridDim.x * blockDim.x;
  for (; i < n; i += stride) p[i] = v;
}

// ------------- WMMA f32 GEMM: C[M,N] = A[M,K] @ W[K,N] + bias -------------
// One wave per 16x64 C strip (4 accumulators): the 16x4 A fragment is loaded
// once per k-step and reused across 4 V_WMMA_F32_16X16X4_F32 issues, so the
// WMMA:VMEM ratio is 4x better than one-tile-per-wave.
// A frag (16x4): lanes 0-15 row M=lane hold K+0,K+1; lanes 16-31 hold K+2,K+3.
// B frag (4x16): mirrored (lane half selects K pair, lane&15 selects N).
// C frag: VGPR j -> row tm*16 + j + (lane>=16 ? 8 : 0), col = base + (lane&15).
__global__ void wmma_gemm_bias(const float* __restrict__ A, const float* __restrict__ W,
                               const float* __restrict__ bias, float* __restrict__ C,
                               int M, int N, int K) {
  const int nt = N >> 6;  // 64-column strips (all N here are multiples of 64)
  const int total = (M >> 4) * nt;
  const int wpb = blockDim.x >> 5;
  int wave = blockIdx.x * wpb + (threadIdx.x >> 5);
  const int lane = threadIdx.x & 31;
  const int stride = gridDim.x * wpb;
  const int half = lane >> 4;
  const int l16 = lane & 15;
  for (int t = wave; t < total; t += stride) {
    const int tm = t / nt, tn = t % nt;
    const float* Arow = A + (long long)(tm * 16 + l16) * K + half * 2;
    const float* Wb = W + tn * 64 + l16 + (long long)(half * 2) * N;
    v8f acc0 = {}, acc1 = {}, acc2 = {}, acc3 = {};
    for (int k0 = 0; k0 < K; k0 += 4) {
      v2f a;
      a.x = Arow[0];
      a.y = Arow[1];
      v2f b0, b1, b2, b3;
      b0.x = Wb[0];      b0.y = Wb[N];
      b1.x = Wb[16];     b1.y = Wb[16 + N];
      b2.x = Wb[32];     b2.y = Wb[32 + N];
      b3.x = Wb[48];     b3.y = Wb[48 + N];
      acc0 = __builtin_amdgcn_wmma_f32_16x16x4_f32(false, a, false, b0, (short)0, acc0, false, false);
      acc1 = __builtin_amdgcn_wmma_f32_16x16x4_f32(false, a, false, b1, (short)0, acc1, false, false);
      acc2 = __builtin_amdgcn_wmma_f32_16x16x4_f32(false, a, false, b2, (short)0, acc2, false, false);
      acc3 = __builtin_amdgcn_wmma_f32_16x16x4_f32(false, a, false, b3, (short)0, acc3, false, false);
      Arow += 4;
      Wb += (long long)4 * N;
    }
    v8f accs[4] = {acc0, acc1, acc2, acc3};
#pragma unroll
    for (int jt = 0; jt < 4; ++jt) {
      const int col = tn * 64 + jt * 16 + l16;
      const float bv = bias ? bias[col] : 0.0f;
      float* Crow = C + (long long)(tm * 16 + half * 8) * N + col;
#pragma unroll
      for (int j = 0; j < 8; ++j) {
        *Crow = accs[jt][j] + bv;
        Crow += N;
      }
    }
  }
}

// ------------- feature-table gather encode -------------
struct TabPtrs { const float* t[9]; };

__global__ void encode_gather_kernel(const int* __restrict__ feat, int nfeat,
                                     TabPtrs tabs, const float* __restrict__ fw,
                                     int ntab, float* __restrict__ out, int M) {
  long long i = (long long)blockIdx.x * blockDim.x + threadIdx.x;
  if (i >= (long long)M * HID) return;
  int r = (int)(i >> 7), c = (int)(i & 127);
  const int* fr = feat + (long long)r * nfeat;
  float acc = 0.0f;
  for (int j = 0; j < ntab; ++j) {
    float s = 1.0f / (1.0f + expf(-fw[j]));
    acc += tabs.t[j][(long long)fr[j] * HID + c] * s;
  }
  out[i] = acc;
}

// ------------- row LayerNorm (+ optional ReLU), wave per row -------------
__global__ void ln_relu_kernel(const float* __restrict__ in, const float* __restrict__ g,
                               const float* __restrict__ b, float* __restrict__ out,
                               int M, int D, int doRelu) {
  int row = blockIdx.x * (blockDim.x >> 5) + (threadIdx.x >> 5);
  int lane = threadIdx.x & 31;
  if (row >= M) return;
  const int per = D >> 5;  // <= 8
  float v[8];
  float s = 0.0f;
  const float* rp = in + (long long)row * D;
  for (int i = 0; i < per; ++i) { v[i] = rp[lane + 32 * i]; s += v[i]; }
  s = waveSum(s);
  float mu = s / (float)D;
  float q = 0.0f;
  for (int i = 0; i < per; ++i) { float d = v[i] - mu; q += d * d; }
  q = waveSum(q);
  float inv = rsqrtf(q / (float)D + 1e-5f);
  float* op = out + (long long)row * D;
  for (int i = 0; i < per; ++i) {
    int c = lane + 32 * i;
    float y = (v[i] - mu) * inv * g[c] + b[c];
    if (doRelu) y = fmaxf(y, 0.0f);
    op[c] = y;
  }
}

// ------------- node encode epilogue: h = relu(LN(pre)) + pos[clip(x[:,2],0,11)] -------------
__global__ void node_post_kernel(const float* __restrict__ pre, const float* __restrict__ g,
                                 const float* __restrict__ b, const int* __restrict__ x,
                                 const float* __restrict__ pos, float* __restrict__ h, int N) {
  int row = blockIdx.x * (blockDim.x >> 5) + (threadIdx.x >> 5);
  int lane = threadIdx.x & 31;
  if (row >= N) return;
  float v[4];
  float s = 0.0f;
  const float* rp = pre + (long long)row * HID;
#pragma unroll
  for (int i = 0; i < 4; ++i) { v[i] = rp[lane + 32 * i]; s += v[i]; }
  s = waveSum(s);
  float mu = s * (1.0f / HID);
  float q = 0.0f;
#pragma unroll
  for (int i = 0; i < 4; ++i) { float d = v[i] - mu; q += d * d; }
  q = waveSum(q);
  float inv = rsqrtf(q * (1.0f / HID) + 1e-5f);
  int deg = x[(long long)row * 9 + 2];
  deg = deg < 0 ? 0 : (deg > 11 ? 11 : deg);
  const float* pr = pos + (long long)deg * HID;
  float* op = h + (long long)row * HID;
#pragma unroll
  for (int i = 0; i < 4; ++i) {
    int c = lane + 32 * i;
    float y = (v[i] - mu) * inv * g[c] + b[c];
    y = fmaxf(y, 0.0f);
    op[c] = y + pr[c];
  }
}

// ------------- layer epilogue: h = relu(LN(attn + S)) + h -------------
__global__ void layer_post_kernel(const float* __restrict__ attn, const float* __restrict__ S,
                                  const float* __restrict__ g, const float* __restrict__ b,
                                  float* __restrict__ h, int N) {
  int row = blockIdx.x * (blockDim.x >> 5) + (threadIdx.x >> 5);
  int lane = threadIdx.x & 31;
  if (row >= N) return;
  float v[4], res[4];
  float s = 0.0f;
  const float* ap = attn + (long long)row * HID;
  const float* sp = S + (long long)row * HID;
  float* op = h + (long long)row * HID;
#pragma unroll
  for (int i = 0; i < 4; ++i) {
    int c = lane + 32 * i;
    v[i] = ap[c] + sp[c];
    res[i] = op[c];
    s += v[i];
  }
  s = waveSum(s);
  float mu = s * (1.0f / HID);
  float q = 0.0f;
#pragma unroll
  for (int i = 0; i < 4; ++i) { float d = v[i] - mu; q += d * d; }
  q = waveSum(q);
  float inv = rsqrtf(q * (1.0f / HID) + 1e-5f);
#pragma unroll
  for (int i = 0; i < 4; ++i) {
    int c = lane + 32 * i;
    float y = (v[i] - mu) * inv * g[c] + b[c];
    op[c] = fmaxf(y, 0.0f) + res[i];
  }
}

// ------------- attention pass A: alpha + segment max -------------
__global__ void edge_attn_alpha(const int* __restrict__ srcI, const int* __restrict__ dstI,
                                const float* __restrict__ Q, const float* __restrict__ Km,
                                const float* __restrict__ Ee, float* __restrict__ alphaB,
                                float* __restrict__ m, int E) {
  int e = blockIdx.x * (blockDim.x >> 5) + (threadIdx.x >> 5);
  int lane = threadIdx.x & 31;
  if (e >= E) return;
  int s = srcI[e], d = dstI[e];
  const float* qr = Q + (long long)d * HID;
  const float* kr = Km + (long long)s * HID;
  const float* er = Ee + (long long)e * HID;
#pragma unroll
  for (int h = 0; h < HEADS; ++h) {
    int c = h * DH + lane;
    float v = qr[c] * (kr[c] + er[c]);
    v = waveSum(v);
    if (lane == 0) {
      float alpha = v * 0.17677669529663687f;  // 1/sqrt(32)
      alphaB[(long long)e * HEADS + h] = alpha;
      atomicMaxF(&m[(long long)d * HEADS + h], alpha);
    }
  }
}

// ------------- attention pass B: exp + segment sum -------------
__global__ void edge_attn_expsum(const int* __restrict__ dstI, const float* __restrict__ m,
                                 float* __restrict__ alphaB, float* __restrict__ ssum, int E) {
  long long i = (long long)blockIdx.x * blockDim.x + threadIdx.x;
  if (i >= (long long)E * HEADS) return;
  int e = (int)(i >> 2), h = (int)(i & 3);
  int d = dstI[e];
  float a = expf(alphaB[i] - m[(long long)d * HEADS + h]);
  alphaB[i] = a;
  atomicAdd(&ssum[(long long)d * HEADS + h], a);
}

// ------------- attention pass C: weighted scatter of V+Ee -------------
__global__ void edge_attn_accum(const int* __restrict__ srcI, const int* __restrict__ dstI,
                                const float* __restrict__ alphaB, const float* __restrict__ ssum,
                                const float* __restrict__ V, const float* __restrict__ Ee,
                                float* __restrict__ attnOut, int E) {
  int e = blockIdx.x * (blockDim.x >> 5) + (threadIdx.x >> 5);
  int lane = threadIdx.x & 31;
  if (e >= E) return;
  int s = srcI[e], d = dstI[e];
  const float* vr = V + (long long)s * HID;
  const float* er = Ee + (long long)e * HID;
  float* orow = attnOut + (long long)d * HID;
#pragma unroll
  for (int h = 0; h < HEADS; ++h) {
    float w = alphaB[(long long)e * HEADS + h] / (ssum[(long long)d * HEADS + h] + 1e-16f);
    int c = h * DH + lane;
    atomicAdd(&orow[c], w * (vr[c] + er[c]));
  }
}

// ------------- pooling -------------
__global__ void pool_logits_kernel(const float* __restrict__ t1, const float* __restrict__ w2,
                                   const float* __restrict__ b2, const int* __restrict__ batch,
                                   float* __restrict__ logits, float* __restrict__ gm, int N) {
  int row = blockIdx.x * (blockDim.x >> 5) + (threadIdx.x >> 5);
  int lane = threadIdx.x & 31;
  if (row >= N) return;
  const float* rp = t1 + (long long)row * HID;
  float s = 0.0f;
#pragma unroll
  for (int i = 0; i < 4; ++i) {
    int c = lane + 32 * i;
    s += tanhf(rp[c]) * w2[c];
  }
  s = waveSum(s);
  if (lane == 0) {
    float L = s + b2[0];
    logits[row] = L;
    atomicMaxF(&gm[batch[row]], L);
  }
}

__global__ void pool_weight_kernel(const int* __restrict__ batch, const float* __restrict__ logits,
                                   const float* __restrict__ gm, float* __restrict__ wout,
                                   float* __restrict__ gz, int N) {
  int i = blockIdx.x * blockDim.x + threadIdx.x;
  if (i >= N) return;
  int b = batch[i];
  float w = expf(logits[i] - gm[b]);
  wout[i] = w;
  atomicAdd(&gz[b], w);
}

__global__ void pool_accum_kernel(const int* __restrict__ batch, const float* __restrict__ h,
                                  const float* __restrict__ w, const float* __restrict__ gz,
                                  float* __restrict__ g, int N) {
  long long i = (long long)blockIdx.x * blockDim.x + threadIdx.x;
  if (i >= (long long)N * HID) return;
  int n = (int)(i >> 7), c = (int)(i & 127);
  int b = batch[n];
  atomicAdd(&g[(long long)b * HID + c], h[i] * (w[n] / (gz[b] + 1e-16f)));
}

// ------------- final row L2 normalize (in place) -------------
__global__ void l2norm_kernel(float* __restrict__ out, int M, int D) {
  int row = blockIdx.x * (blockDim.x >> 5) + (threadIdx.x >> 5);
  int lane = threadIdx.x & 31;
  if (row >= M) return;
  const int per = D >> 5;
  float v[8];
  float s = 0.0f;
  float* rp = out + (long long)row * D;
  for (int i = 0; i < per; ++i) { v[i] = rp[lane + 32 * i]; s += v[i] * v[i]; }
  s = waveSum(s);
  float inv = 1.0f / fmaxf(sqrtf(s), 1e-12f);
  for (int i = 0; i < per; ++i) rp[lane + 32 * i] = v[i] * inv;
}

// =====================================================================
// Host side
// Input flattening assumed (setup_inputs() dict insertion order, recursive):
//  0 x[N,9] 1 edge_attr[E,3] 2 edge_index[2,E] 3 batch[N]
//  atom: 4..12 tables, 13 fw, 14 combine.w, 15 combine.b, 16 ln_g, 17 ln_b
//  bond: 18..20 tables, 21 fw, 22 combine.w, 23 combine.b, 24 ln_g, 25 ln_b
//  26 pos[12,128]
//  layer L (base 27+12L): qw qb kw kb vw vb ew eb skipw skipb ln_g ln_b
//  75 pool.l1.w 76 pool.l1.b 77 pool.l2.w 78 pool.l2.b
//  79 proj.l1.w 80 proj.l1.b 81 ln1_g 82 ln1_b 83 l2.w 84 l2.b 85 ln2_g 86 ln2_b
//  87 l3.w 88 l3.b
// =====================================================================
extern "C" void kernel_launch(void* const* d_in, const int* in_sizes, int n_in,
                              void* d_out, int out_size, void* d_ws, size_t ws_size,
                              hipStream_t stream) {
  const int* x = (const int*)d_in[0];
  const int* edge_index = (const int*)d_in[2];
  const int* batch = (const int*)d_in[3];
  const int N = in_sizes[0] / 9;
  const int E = in_sizes[1] / 3;
  const int* srcI = edge_index;
  const int* dstI = edge_index + E;
  auto F = [&](int i) -> const float* { return (const float*)d_in[i]; };

  float* ws = (float*)d_ws;
  long long off = 0;
  auto alloc = [&](long long n) { float* p = ws + off; off += n; return p; };
  float* hB    = alloc((long long)N * HID);
  float* preB  = alloc((long long)N * HID);  // encode pre-LN / attnOut
  float* Qb    = alloc((long long)N * HID);  // also gather tmp / poolTmp
  float* Kb    = alloc((long long)N * HID);
  float* Vb    = alloc((long long)N * HID);
  float* Sb    = alloc((long long)N * HID);
  float* eEnc  = alloc((long long)E * HID);
  float* eTmp  = alloc((long long)E * HID);  // bond gather/pre, then Ee per layer
  float* alphaB = alloc((long long)E * HEADS);
  float* mB    = alloc((long long)N * HEADS);
  float* sB    = alloc((long long)N * HEADS);
  float* logitsB = alloc(N);
  float* wB    = alloc(N);
  float* gmB   = alloc(NGRAPH);
  float* gzB   = alloc(NGRAPH);
  float* gB    = alloc((long long)NGRAPH * HID);
  float* p1    = alloc((long long)NGRAPH * 2 * HID);  // 512x256
  (void)ws_size; (void)n_in;

  const dim3 blk(256);
  auto gemm = [&](const float* A, const float* W, const float* bias, float* C,
                  int M_, int N_, int K_) {
    int tiles = (M_ >> 4) * (N_ >> 6);  // 16x64 strips per wave
    int blocks = (tiles + 7) / 8;
    if (blocks < 1) blocks = 1;
    wmma_gemm_bias<<<dim3(blocks), blk, 0, stream>>>(A, W, bias, C, M_, N_, K_);
  };
  auto fill = [&](float* p, float v, long long n) {
    fill_kernel<<<dim3(4096), blk, 0, stream>>>(p, v, n);
  };

  // ---- atom encode ----
  TabPtrs at{};
  for (int j = 0; j < 9; ++j) at.t[j] = F(4 + j);
  {
    long long tot = (long long)N * HID;
    encode_gather_kernel<<<dim3((unsigned)((tot + 255) / 256)), blk, 0, stream>>>(
        x, 9, at, F(13), 9, Qb, N);
  }
  gemm(Qb, F(14), F(15), preB, N, HID, HID);
  node_post_kernel<<<dim3((N + 7) / 8), blk, 0, stream>>>(preB, F(16), F(17), x, F(26), hB, N);

  // ---- bond encode ----
  TabPtrs bt{};
  for (int j = 0; j < 3; ++j) bt.t[j] = F(18 + j);
  {
    long long tot = (long long)E * HID;
    encode_gather_kernel<<<dim3((unsigned)((tot + 255) / 256)), blk, 0, stream>>>(
        (const int*)d_in[1], 3, bt, F(21), 3, eTmp, E);
  }
  gemm(eTmp, F(22), F(23), eEnc, E, HID, HID);
  ln_relu_kernel<<<dim3((E + 7) / 8), blk, 0, stream>>>(eEnc, F(24), F(25), eEnc, E, HID, 1);

  // ---- transformer layers ----
  for (int L = 0; L < 4; ++L) {
    int BL = 27 + 12 * L;
    gemm(hB, F(BL + 0), F(BL + 1), Qb, N, HID, HID);      // Q
    gemm(hB, F(BL + 2), F(BL + 3), Kb, N, HID, HID);      // K
    gemm(hB, F(BL + 4), F(BL + 5), Vb, N, HID, HID);      // V
    gemm(eEnc, F(BL + 6), F(BL + 7), eTmp, E, HID, HID);  // Ee
    gemm(hB, F(BL + 8), F(BL + 9), Sb, N, HID, HID);      // skip

    fill(mB, -INFINITY, (long long)N * HEADS);
    fill(sB, 0.0f, (long long)N * HEADS);
    fill(preB, 0.0f, (long long)N * HID);

    edge_attn_alpha<<<dim3((E + 7) / 8), blk, 0, stream>>>(srcI, dstI, Qb, Kb, eTmp,
                                                           alphaB, mB, E);
    {
      long long tot = (long long)E * HEADS;
      edge_attn_expsum<<<dim3((unsigned)((tot + 255) / 256)), blk, 0, stream>>>(
          dstI, mB, alphaB, sB, E);
    }
    edge_attn_accum<<<dim3((E + 7) / 8), blk, 0, stream>>>(srcI, dstI, alphaB, sB, Vb,
                                                           eTmp, preB, E);
    layer_post_kernel<<<dim3((N + 7) / 8), blk, 0, stream>>>(preB, Sb, F(BL + 10),
                                                             F(BL + 11), hB, N);
  }

  // ---- attention pooling ----
  gemm(hB, F(75), F(76), Qb, N, HID, HID);  // pool.l1
  fill(gmB, -INFINITY, NGRAPH);
  fill(gzB, 0.0f, NGRAPH);
  fill(gB, 0.0f, (long long)NGRAPH * HID);
  pool_logits_kernel<<<dim3((N + 7) / 8), blk, 0, stream>>>(Qb, F(77), F(78), batch,
                                                            logitsB, gmB, N);
  pool_weight_kernel<<<dim3((N + 255) / 256), blk, 0, stream>>>(batch, logitsB, gmB, wB,
                                                                gzB, N);
  {
    long long tot = (long long)N * HID;
    pool_accum_kernel<<<dim3((unsigned)((tot + 255) / 256)), blk, 0, stream>>>(
        batch, hB, wB, gzB, gB, N);
  }

  // ---- projection head ----
  float* outF = (float*)d_out;
  gemm(gB, F(79), F(80), p1, NGRAPH, 2 * HID, HID);
  ln_relu_kernel<<<dim3((NGRAPH + 7) / 8), blk, 0, stream>>>(p1, F(81), F(82), p1,
                                                             NGRAPH, 2 * HID, 1);
  gemm(p1, F(83), F(84), gB, NGRAPH, HID, 2 * HID);
  ln_relu_kernel<<<dim3((NGRAPH + 7) / 8), blk, 0, stream>>>(gB, F(85), F(86), gB,
                                                             NGRAPH, HID, 1);
  gemm(gB, F(87), F(88), outF, NGRAPH, 2 * HID, HID);
  l2norm_kernel<<<dim3((NGRAPH + 7) / 8), blk, 0, stream>>>(outF, NGRAPH, 2 * HID);
  (void)out_size;
}